// Convolution_54563264528556
// MI455X (gfx1250) — compile-verified
//
#include <hip/hip_runtime.h>
#include <math.h>

#define N_NODES 20000
#define N_EDGES 400000

typedef float v2f __attribute__((ext_vector_type(2)));
typedef float v8f __attribute__((ext_vector_type(8)));

// folded scale constants
#define S_LIN 0.25f                      // 1/sqrt(16)
#define S_H0  0.25f                      // 1/sqrt(16)  (MLP layer0)
#define S_H   0.35355339059327373f       // 1/sqrt(8)   (MLP hidden)
#define S_K3  0.04419417382415922f       // 1/(sqrt(8)*8)   = w3 scale * 1/sqrt(64)
#define S_V3  0.17677669529663687f       // 1/(sqrt(8)*2)   = w3 scale * 1/sqrt(4)
#define S_SC  0.08838834764831845f       // 1/sqrt(128)
#define S_DOT 0.125f                     // 1/sqrt(64)

__device__ __forceinline__ float silu_f(float x) { return x / (1.0f + expf(-x)); }

__device__ __forceinline__ unsigned ord_enc(float f) {
  unsigned b = __float_as_uint(f);
  return (b & 0x80000000u) ? ~b : (b | 0x80000000u);
}
__device__ __forceinline__ float ord_dec(unsigned u) {
  unsigned b = (u & 0x80000000u) ? (u ^ 0x80000000u) : ~u;
  return __uint_as_float(b);
}

// ---------------------------------------------------------------------------
// K1: per-node prework. sc, nf, q; also zero z/agg and init global max.
// ---------------------------------------------------------------------------
__global__ __launch_bounds__(256) void node_pre_kernel(
    const float* __restrict__ node_input, const float* __restrict__ node_attr,
    const float* __restrict__ W_sc, const float* __restrict__ W_lin1,
    const float* __restrict__ W_hq,
    float* __restrict__ nf, float* __restrict__ q, float* __restrict__ sc,
    float* __restrict__ z, float* __restrict__ agg, unsigned* __restrict__ maxu) {
  __shared__ float sWsc[2048];
  __shared__ float sW1[256];
  __shared__ float sWq[128];
  const int tid = threadIdx.x;
  for (int i = tid; i < 2048; i += 256) sWsc[i] = W_sc[i] * S_SC;
  sW1[tid] = W_lin1[tid] * S_LIN;
  if (tid < 128) sWq[tid] = W_hq[tid] * S_LIN;
  __syncthreads();
  if (blockIdx.x == 0 && tid == 0) *maxu = 0u;

  const int n = blockIdx.x * 256 + tid;
  if (n >= N_NODES) return;

  float in[16], at[8];
  {
    const float4* p = (const float4*)(node_input + n * 16);
    float4 a = p[0], b = p[1], c = p[2], d = p[3];
    in[0]=a.x; in[1]=a.y; in[2]=a.z; in[3]=a.w;
    in[4]=b.x; in[5]=b.y; in[6]=b.z; in[7]=b.w;
    in[8]=c.x; in[9]=c.y; in[10]=c.z; in[11]=c.w;
    in[12]=d.x; in[13]=d.y; in[14]=d.z; in[15]=d.w;
    const float4* pa = (const float4*)(node_attr + n * 8);
    float4 e = pa[0], f = pa[1];
    at[0]=e.x; at[1]=e.y; at[2]=e.z; at[3]=e.w;
    at[4]=f.x; at[5]=f.y; at[6]=f.z; at[7]=f.w;
  }
  // nf = node_input @ (W_lin1/4)
  float nfv[16];
#pragma unroll
  for (int j = 0; j < 16; ++j) {
    float acc = 0.f;
#pragma unroll
    for (int i = 0; i < 16; ++i) acc += in[i] * sW1[i * 16 + j];
    nfv[j] = acc;
    nf[n * 16 + j] = acc;
  }
  // q = nf @ (W_hq/4)
#pragma unroll
  for (int j = 0; j < 8; ++j) {
    float acc = 0.f;
#pragma unroll
    for (int i = 0; i < 16; ++i) acc += nfv[i] * sWq[i * 8 + j];
    q[n * 8 + j] = acc;
  }
  // sc[c] = sum_{a,b} in[a]*at[b]*Wsc[a,b,c] (scale folded)
  float s[16];
#pragma unroll
  for (int c = 0; c < 16; ++c) s[c] = 0.f;
#pragma unroll
  for (int a = 0; a < 16; ++a) {
#pragma unroll
    for (int b = 0; b < 8; ++b) {
      float t = in[a] * at[b];
      const float* w = sWsc + (a * 8 + b) * 16;
#pragma unroll
      for (int c = 0; c < 16; ++c) s[c] += t * w[c];
    }
  }
#pragma unroll
  for (int c = 0; c < 16; ++c) sc[n * 16 + c] = s[c];
  z[n] = 0.f;
#pragma unroll
  for (int u = 0; u < 16; ++u) agg[n * 16 + u] = 0.f;
}

// ---------------------------------------------------------------------------
// K2: heavy edge kernel. One wave = 16 edges. WMMA f32 16x16x4 GEMMs.
// B matrices stored K-pair-interleaved in LDS so each WMMA B operand is a
// single aligned ds_load_b64 straight into an even VGPR pair.
// ---------------------------------------------------------------------------
#define K2_WAVES 4
#define K2_THREADS (K2_WAVES * 32)

__global__ __launch_bounds__(K2_THREADS) void edge_kernel(
    const int* __restrict__ edge_src, const int* __restrict__ edge_dst,
    const float* __restrict__ edge_attr, const float* __restrict__ edge_scalars,
    const float* __restrict__ fck_w0, const float* __restrict__ fck_w1,
    const float* __restrict__ fck_w2, const float* __restrict__ fck_w3,
    const float* __restrict__ fcv_w0, const float* __restrict__ fcv_w1,
    const float* __restrict__ fcv_w2, const float* __restrict__ fcv_w3,
    const float* __restrict__ W_dot,
    const float* __restrict__ nf, const float* __restrict__ q,
    float* __restrict__ Xbuf, float* __restrict__ Vbuf,
    unsigned* __restrict__ maxu) {
  // shared: W3rI 4096 | W3vI 512 | Wk0 128 | Wk1 64 | Wk2 64 | Wv0 128 | Wv1 64
  //         Wv2 64 | Wdot 64   then per-wave: SF 256 | EA 64 | SCR 2048
  __shared__ __align__(16) float smem[5184 + K2_WAVES * 2368];
  float* sW3rI = smem;                // interleaved: [(k/2)*64+n]*2 + (k&1)
  float* sW3vI = smem + 4096;         // interleaved: [(v/2)*128+n]*2 + (v&1)
  float* sWk0 = smem + 4608;
  float* sWk1 = smem + 4736;
  float* sWk2 = smem + 4800;
  float* sWv0 = smem + 4864;
  float* sWv1 = smem + 4992;
  float* sWv2 = smem + 5056;
  float* sWdot = smem + 5120;

  const int tid = threadIdx.x;
  // Stage + reshape + scale constant weights into LDS (hits L2 across blocks).
  for (int i = tid; i < 4096; i += K2_THREADS) {
    int j = i & 1, t = i >> 1;
    int n = t & 63, p = t >> 6;
    int k = 2 * p + j;                 // uv index 0..63
    int u = k >> 2, v = k & 3, h = n >> 3, c = n & 7;
    sW3rI[i] = fck_w3[h * 512 + u * 32 + v * 8 + c] * S_K3;
  }
  for (int i = tid; i < 512; i += K2_THREADS) {
    int j = i & 1, t = i >> 1;
    int n = t & 127, p = t >> 7;
    int v = 2 * p + j;                 // v index 0..3
    int h = n >> 4, u = n & 15;
    sW3vI[i] = fcv_w3[h * 64 + u * 4 + v] * S_V3;
  }
  if (tid < 128) { sWk0[tid] = fck_w0[tid] * S_H0; sWv0[tid] = fcv_w0[tid] * S_H0; }
  if (tid < 64) {
    sWk1[tid] = fck_w1[tid] * S_H;  sWk2[tid] = fck_w2[tid] * S_H;
    sWv1[tid] = fcv_w1[tid] * S_H;  sWv2[tid] = fcv_w2[tid] * S_H;
    sWdot[tid] = W_dot[tid] * S_DOT;
  }
  __syncthreads();

  const int wave = tid >> 5;
  const int lane = tid & 31;
  const int row = lane & 15;   // edge-in-tile for A/C rows, N-col for B
  const int hi = lane >> 4;    // K-half selector per ISA f32 A/B layouts

  float* sSF = smem + 5184 + wave * 2368;   // [16][16] gathered src_f
  float* sEA = sSF + 256;                   // [16][4]  edge_attr
  float* sSCR = sEA + 64;                   // 2048: B2 [16][64] then Cv [16][128]

  const int e0 = (blockIdx.x * K2_WAVES + wave) * 16;
  const int eMy = e0 + row;

  // --- gather stage: half-wave split ---
  if (hi == 0) {
    int isrc = edge_src[eMy];
    const float4* p = (const float4*)(nf + isrc * 16);
    float4* d = (float4*)(sSF + row * 16);
    d[0] = p[0]; d[1] = p[1]; d[2] = p[2]; d[3] = p[3];
  } else {
    *(float4*)(sEA + row * 4) = *(const float4*)(edge_attr + eMy * 4);
  }

  // --- per-edge MLP hiddens: lanes<16 -> fc_k, lanes>=16 -> fc_v (same edges)
  float es[16];
  {
    const float4* p = (const float4*)(edge_scalars + eMy * 16);
    float4 a = p[0], b = p[1], c = p[2], d = p[3];
    es[0]=a.x; es[1]=a.y; es[2]=a.z; es[3]=a.w;
    es[4]=b.x; es[5]=b.y; es[6]=b.z; es[7]=b.w;
    es[8]=c.x; es[9]=c.y; es[10]=c.z; es[11]=c.w;
    es[12]=d.x; es[13]=d.y; es[14]=d.z; es[15]=d.w;
  }
  const float* W0 = hi ? sWv0 : sWk0;
  const float* W1 = hi ? sWv1 : sWk1;
  const float* W2 = hi ? sWv2 : sWk2;
  float h0[8], h1[8], h2[8];
#pragma unroll
  for (int j = 0; j < 8; ++j) {
    float acc = 0.f;
#pragma unroll
    for (int i = 0; i < 16; ++i) acc += es[i] * W0[i * 8 + j];
    h0[j] = silu_f(acc);
  }
#pragma unroll
  for (int j = 0; j < 8; ++j) {
    float acc = 0.f;
#pragma unroll
    for (int i = 0; i < 8; ++i) acc += h0[i] * W1[i * 8 + j];
    h1[j] = silu_f(acc);
  }
#pragma unroll
  for (int j = 0; j < 8; ++j) {
    float acc = 0.f;
#pragma unroll
    for (int i = 0; i < 8; ++i) acc += h1[i] * W2[i * 8 + j];
    h2[j] = silu_f(acc);   // stays in registers; consumed by E-k / E-v below
  }

  // --- GEMM1: B2[16e x 64hc] = O[16 x 64] @ W3r[64 x 64], O = SF (x) EA ---
  // f32 16x16x4 A layout: lanes0-15 K0/K1, lanes16-31 K2/K3.
  const float ea0 = sEA[row * 4 + 2 * hi];
  const float ea1 = sEA[row * 4 + 2 * hi + 1];
  v8f acc0 = {}, acc1 = {}, acc2 = {}, acc3 = {};
#pragma unroll
  for (int kg = 0; kg < 4; ++kg) {
    float4 sf4 = *(const float4*)(sSF + row * 16 + kg * 4);
    float sfe[4] = {sf4.x, sf4.y, sf4.z, sf4.w};
#pragma unroll
    for (int ki = 0; ki < 4; ++ki) {
      const int kk = kg * 4 + ki;
      const float sf = sfe[ki];
      v2f a; a.x = sf * ea0; a.y = sf * ea1;
      // interleaved pair index p = (4kk+2hi)/2 = 2kk+hi
      const float* bp = sW3rI + 2 * ((2 * kk + hi) * 64 + row);
      v2f b0 = *(const v2f*)(bp);
      v2f b1 = *(const v2f*)(bp + 32);
      v2f b2 = *(const v2f*)(bp + 64);
      v2f b3 = *(const v2f*)(bp + 96);
      acc0 = __builtin_amdgcn_wmma_f32_16x16x4_f32(false, a, false, b0, (short)0, acc0, false, false);
      acc1 = __builtin_amdgcn_wmma_f32_16x16x4_f32(false, a, false, b1, (short)0, acc1, false, false);
      acc2 = __builtin_amdgcn_wmma_f32_16x16x4_f32(false, a, false, b2, (short)0, acc2, false, false);
      acc3 = __builtin_amdgcn_wmma_f32_16x16x4_f32(false, a, false, b3, (short)0, acc3, false, false);
    }
  }
  // dump C tiles: VGPR r, lanes0-15 -> M=r, lanes16-31 -> M=r+8; N = lane%16
#pragma unroll
  for (int r = 0; r < 8; ++r) {
    const int m = r + 8 * hi;
    sSCR[m * 64 + row]      = acc0[r];
    sSCR[m * 64 + 16 + row] = acc1[r];
    sSCR[m * 64 + 32 + row] = acc2[r];
    sSCR[m * 64 + 48 + row] = acc3[r];
  }

  // --- E-k: lanes 0-15 finish k and attention logit for their edge ---
  float xlog = -__builtin_huge_valf();
  if (hi == 0) {
    float kv[8];
#pragma unroll
    for (int c = 0; c < 8; ++c) kv[c] = 0.f;
#pragma unroll
    for (int h = 0; h < 8; ++h) {
      const float hh = h2[h];
      float4 s0 = *(const float4*)(sSCR + row * 64 + h * 8);
      float4 s1 = *(const float4*)(sSCR + row * 64 + h * 8 + 4);
      kv[0] += hh * s0.x; kv[1] += hh * s0.y; kv[2] += hh * s0.z; kv[3] += hh * s0.w;
      kv[4] += hh * s1.x; kv[5] += hh * s1.y; kv[6] += hh * s1.z; kv[7] += hh * s1.w;
    }
    const int dst = edge_dst[eMy];
    const float* qp = q + dst * 8;
    float qd[8];
#pragma unroll
    for (int a = 0; a < 8; ++a) qd[a] = qp[a];
    float x = 0.f;
#pragma unroll
    for (int b = 0; b < 8; ++b) {
      float t = 0.f;
#pragma unroll
      for (int a = 0; a < 8; ++a) t += qd[a] * sWdot[a * 8 + b];
      x += t * kv[b];
    }
    Xbuf[eMy] = x;
    xlog = x;
  }
  // wave-level max reduce, 1 atomic per wave
#pragma unroll
  for (int off = 16; off > 0; off >>= 1) xlog = fmaxf(xlog, __shfl_xor(xlog, off, 32));
  if (lane == 0) atomicMax(maxu, ord_enc(xlog));

  // --- GEMM2: Cv[16e x 128(h,u)] = EA[16 x 4] @ W3v[4 x 128] ---
  v2f a2; a2.x = ea0; a2.y = ea1;
  v8f vacc[8];
  {
    const float* bvp = sW3vI + 2 * (hi * 128 + row);
#pragma unroll
    for (int nt = 0; nt < 8; ++nt) {
      v2f b = *(const v2f*)(bvp + 32 * nt);
      v8f cz = {};
      vacc[nt] = __builtin_amdgcn_wmma_f32_16x16x4_f32(false, a2, false, b, (short)0, cz, false, false);
    }
  }
#pragma unroll
  for (int nt = 0; nt < 8; ++nt) {
#pragma unroll
    for (int r = 0; r < 8; ++r) sSCR[(r + 8 * hi) * 128 + nt * 16 + row] = vacc[nt][r];
  }

  // --- E-v: lanes 16-31 finish v for their edge ---
  if (hi == 1) {
    float av[16];
#pragma unroll
    for (int u = 0; u < 16; ++u) av[u] = 0.f;
#pragma unroll
    for (int h = 0; h < 8; ++h) {
      const float hh = h2[h];
#pragma unroll
      for (int jj = 0; jj < 4; ++jj) {
        float4 s = *(const float4*)(sSCR + row * 128 + h * 16 + jj * 4);
        av[jj * 4 + 0] += hh * s.x; av[jj * 4 + 1] += hh * s.y;
        av[jj * 4 + 2] += hh * s.z; av[jj * 4 + 3] += hh * s.w;
      }
    }
    float vv[16];
#pragma unroll
    for (int u = 0; u < 16; ++u) vv[u] = sSF[row * 16 + u] * av[u];
    float4* vp = (float4*)(Vbuf + eMy * 16);
    vp[0] = make_float4(vv[0], vv[1], vv[2], vv[3]);
    vp[1] = make_float4(vv[4], vv[5], vv[6], vv[7]);
    vp[2] = make_float4(vv[8], vv[9], vv[10], vv[11]);
    vp[3] = make_float4(vv[12], vv[13], vv[14], vv[15]);
  }
}

// ---------------------------------------------------------------------------
// K3: softmax numerator + scatter sums (float global atomics -> L2)
// ---------------------------------------------------------------------------
__global__ __launch_bounds__(256) void edge_soft_kernel(
    const int* __restrict__ edge_dst, const float* __restrict__ Xbuf,
    const float* __restrict__ Vbuf, const unsigned* __restrict__ maxu,
    float* __restrict__ z, float* __restrict__ agg) {
  const int e = blockIdx.x * 256 + threadIdx.x;
  if (e >= N_EDGES) return;
  const float gmax = ord_dec(*maxu);
  const float ex = expf(Xbuf[e] - gmax);
  const int dst = edge_dst[e];
  atomicAdd(z + dst, ex);
  const float4* vp = (const float4*)(Vbuf + e * 16);
  float4 v0 = vp[0], v1 = vp[1], v2 = vp[2], v3 = vp[3];
  float* ap = agg + dst * 16;
  atomicAdd(ap + 0,  ex * v0.x); atomicAdd(ap + 1,  ex * v0.y);
  atomicAdd(ap + 2,  ex * v0.z); atomicAdd(ap + 3,  ex * v0.w);
  atomicAdd(ap + 4,  ex * v1.x); atomicAdd(ap + 5,  ex * v1.y);
  atomicAdd(ap + 6,  ex * v1.z); atomicAdd(ap + 7,  ex * v1.w);
  atomicAdd(ap + 8,  ex * v2.x); atomicAdd(ap + 9,  ex * v2.y);
  atomicAdd(ap + 10, ex * v2.z); atomicAdd(ap + 11, ex * v2.w);
  atomicAdd(ap + 12, ex * v3.x); atomicAdd(ap + 13, ex * v3.y);
  atomicAdd(ap + 14, ex * v3.z); atomicAdd(ap + 15, ex * v3.w);
}

// ---------------------------------------------------------------------------
// K4: out = sc + (agg/z) @ (W_lin2/4)
// ---------------------------------------------------------------------------
__global__ __launch_bounds__(256) void node_out_kernel(
    const float* __restrict__ sc, const float* __restrict__ z,
    const float* __restrict__ agg, const float* __restrict__ W_lin2,
    float* __restrict__ out) {
  __shared__ float sW2[256];
  sW2[threadIdx.x] = W_lin2[threadIdx.x] * S_LIN;
  __syncthreads();
  const int n = blockIdx.x * 256 + threadIdx.x;
  if (n >= N_NODES) return;
  float zz = z[n];
  if (zz == 0.f) zz = 1.f;
  const float inv = 1.f / zz;
  float a[16];
#pragma unroll
  for (int u = 0; u < 16; ++u) a[u] = agg[n * 16 + u] * inv;
  float4* op = (float4*)(out + n * 16);
  float o[16];
#pragma unroll
  for (int c = 0; c < 16; ++c) {
    float acc = sc[n * 16 + c];
#pragma unroll
    for (int u = 0; u < 16; ++u) acc += a[u] * sW2[u * 16 + c];
    o[c] = acc;
  }
  op[0] = make_float4(o[0], o[1], o[2], o[3]);
  op[1] = make_float4(o[4], o[5], o[6], o[7]);
  op[2] = make_float4(o[8], o[9], o[10], o[11]);
  op[3] = make_float4(o[12], o[13], o[14], o[15]);
}

// ---------------------------------------------------------------------------
extern "C" void kernel_launch(void* const* d_in, const int* in_sizes, int n_in,
                              void* d_out, int out_size, void* d_ws, size_t ws_size,
                              hipStream_t stream) {
  (void)in_sizes; (void)n_in; (void)out_size; (void)ws_size;
  const float* node_input   = (const float*)d_in[0];
  const float* node_attr    = (const float*)d_in[1];
  const int*   edge_src     = (const int*)d_in[2];
  const int*   edge_dst     = (const int*)d_in[3];
  const float* edge_attr    = (const float*)d_in[4];
  const float* edge_scalars = (const float*)d_in[5];
  const float* W_sc    = (const float*)d_in[6];
  const float* W_lin1  = (const float*)d_in[7];
  const float* W_hq    = (const float*)d_in[8];
  const float* fck_w0  = (const float*)d_in[9];
  const float* fck_w1  = (const float*)d_in[10];
  const float* fck_w2  = (const float*)d_in[11];
  const float* fck_w3  = (const float*)d_in[12];
  const float* fcv_w0  = (const float*)d_in[13];
  const float* fcv_w1  = (const float*)d_in[14];
  const float* fcv_w2  = (const float*)d_in[15];
  const float* fcv_w3  = (const float*)d_in[16];
  const float* W_dot   = (const float*)d_in[17];
  const float* W_lin2  = (const float*)d_in[18];

  float* ws   = (float*)d_ws;
  float* nf   = ws;                 // N*16
  float* q    = ws + 320000;        // N*8
  float* sc   = ws + 480000;        // N*16
  float* z    = ws + 800000;        // N
  float* agg  = ws + 820000;        // N*16
  float* Xbuf = ws + 1140000;       // E
  float* Vbuf = ws + 1540000;       // E*16
  unsigned* maxu = (unsigned*)(ws + 7940000);
  float* out  = (float*)d_out;

  node_pre_kernel<<<(N_NODES + 255) / 256, 256, 0, stream>>>(
      node_input, node_attr, W_sc, W_lin1, W_hq, nf, q, sc, z, agg, maxu);

  edge_kernel<<<N_EDGES / (K2_WAVES * 16), K2_THREADS, 0, stream>>>(
      edge_src, edge_dst, edge_attr, edge_scalars,
      fck_w0, fck_w1, fck_w2, fck_w3,
      fcv_w0, fcv_w1, fcv_w2, fcv_w3,
      W_dot, nf, q, Xbuf, Vbuf, maxu);

  edge_soft_kernel<<<(N_EDGES + 255) / 256, 256, 0, stream>>>(
      edge_dst, Xbuf, Vbuf, maxu, z, agg);

  node_out_kernel<<<(N_NODES + 255) / 256, 256, 0, stream>>>(
      sc, z, agg, W_lin2, out);
}